// PSRoIAlign_10591389352594
// MI455X (gfx1250) — compile-verified
//
#include <hip/hip_runtime.h>
#include <stdint.h>

// Problem constants from the reference.
#define PS_C      490
#define PS_H      100
#define PS_W      100
#define PS_POOL   7
#define PS_NSAMP  14          // POOL_H * GRID distinct axis samples per ROI
#define PS_ELEMS  490         // C_out(10) * 7 * 7 outputs per ROI; channel == element index
#define PS_SCALE  0.0625f

// Matches the reference _axis_interp exactly.
__device__ __forceinline__ void axis_interp(float t, int size,
                                            int& lo, int& hi,
                                            float& wl, float& wh, int& valid) {
    valid = (t >= -1.0f) && (t <= (float)size);
    float tc = fmaxf(t, 0.0f);
    int low = (int)floorf(tc);
    low = min(low, size - 1);
    hi = min(low + 1, size - 1);
    float frac = (low >= size - 1) ? 0.0f : (tc - (float)low);
    lo = low;
    wl = 1.0f - frac;
    wh = frac;
}

__global__ __launch_bounds__(256) void psroialign_gfx1250_kernel(
        const float* __restrict__ input,   // (4, 490, 100, 100)
        const float* __restrict__ rois,    // (K, 5)
        float* __restrict__ out) {         // (K, 10, 7, 7)
    const int tid = threadIdx.x;
    const int k   = blockIdx.x;

    __shared__ float s_roi[8];
    __shared__ float s_wyl[PS_NSAMP], s_wyh[PS_NSAMP], s_wxl[PS_NSAMP], s_wxh[PS_NSAMP];
    __shared__ int   s_yl[PS_NSAMP], s_yh[PS_NSAMP], s_xl[PS_NSAMP], s_xh[PS_NSAMP];
    __shared__ int   s_vy[PS_NSAMP], s_vx[PS_NSAMP];
    __shared__ int   s_b;

    // ---- Stage the 5 ROI floats into LDS via the CDNA5 async global->LDS DMA.
    // Lanes 0..4 (wave 0) each copy one dword; ASYNCcnt tracks completion.
    if (tid < 5) {
        const float* gptr = rois + (size_t)k * 5 + tid;
        uint32_t lds_addr = (uint32_t)(uintptr_t)(&s_roi[tid]);  // low 32 bits = LDS byte offset
        uint64_t gaddr    = (uint64_t)(uintptr_t)gptr;
        asm volatile("global_load_async_to_lds_b32 %0, %1, off"
                     :: "v"(lds_addr), "v"(gaddr)
                     : "memory");
    }
    asm volatile("s_wait_asynccnt 0x0" ::: "memory");
    __syncthreads();

    // ---- Per-ROI axis interpolation pieces (14 y-samples, 14 x-samples) into LDS.
    if (tid < PS_NSAMP) {
        // y axis: sample s = ph*2 + g
        int s  = tid;
        int ph = s >> 1;
        int g  = s & 1;
        float y1  = s_roi[2] * PS_SCALE - 0.5f;
        float y2  = s_roi[4] * PS_SCALE - 0.5f;
        float bsh = (y2 - y1) * (1.0f / (float)PS_POOL);
        float y   = y1 + ((float)ph + ((float)g + 0.5f) * 0.5f) * bsh;
        axis_interp(y, PS_H, s_yl[s], s_yh[s], s_wyl[s], s_wyh[s], s_vy[s]);
    } else if (tid >= 32 && tid < 32 + PS_NSAMP) {
        // x axis on a separate wave (wave 1) to keep lanes busy
        int s  = tid - 32;
        int pw = s >> 1;
        int g  = s & 1;
        float x1  = s_roi[1] * PS_SCALE - 0.5f;
        float x2  = s_roi[3] * PS_SCALE - 0.5f;
        float bsw = (x2 - x1) * (1.0f / (float)PS_POOL);
        float x   = x1 + ((float)pw + ((float)g + 0.5f) * 0.5f) * bsw;
        axis_interp(x, PS_W, s_xl[s], s_xh[s], s_wxl[s], s_wxh[s], s_vx[s]);
    } else if (tid == 64) {
        s_b = (int)s_roi[0];
    }
    __syncthreads();

    // ---- Gather + accumulate. Output element e of ROI k reads channel c == e.
    const float* in_n = input + (size_t)s_b * PS_C * PS_H * PS_W;
    float* out_k = out + (size_t)k * PS_ELEMS;

    for (int e = tid; e < PS_ELEMS; e += 256) {
        int bin = e % 49;            // ph*7 + pw
        int ph  = bin / 7;
        int pw  = bin - ph * 7;
        const float* plane = in_n + (size_t)e * (PS_H * PS_W);

        float acc = 0.0f;
        #pragma unroll
        for (int iy = 0; iy < 2; ++iy) {
            int ys = ph * 2 + iy;
            if (!s_vy[ys]) continue;
            const float wyl = s_wyl[ys], wyh = s_wyh[ys];
            const float* rowl = plane + s_yl[ys] * PS_W;
            const float* rowh = plane + s_yh[ys] * PS_W;
            #pragma unroll
            for (int ix = 0; ix < 2; ++ix) {
                int xs = pw * 2 + ix;
                if (!s_vx[xs]) continue;
                const int   xl  = s_xl[xs], xh = s_xh[xs];
                const float wxl = s_wxl[xs], wxh = s_wxh[xs];
                acc += wyl * (wxl * rowl[xl] + wxh * rowl[xh])
                     + wyh * (wxl * rowh[xl] + wxh * rowh[xh]);
            }
        }
        out_k[e] = acc * 0.25f;   // average over GRID*GRID samples
    }
}

extern "C" void kernel_launch(void* const* d_in, const int* in_sizes, int n_in,
                              void* d_out, int out_size, void* d_ws, size_t ws_size,
                              hipStream_t stream) {
    const float* input = (const float*)d_in[0];
    const float* rois  = (const float*)d_in[1];
    float* out = (float*)d_out;
    const int K = in_sizes[1] / 5;   // rois is (K, 5)

    psroialign_gfx1250_kernel<<<dim3(K), dim3(256), 0, stream>>>(input, rois, out);
}